// SoftcbDiceLoss_85856396247138
// MI455X (gfx1250) — compile-verified
//
#include <hip/hip_runtime.h>
#include <math.h>

// ---------------------------------------------------------------------------
// SoftcbDiceLoss (1,1,96,128,128) fp32 -> scalar.
// Fused soft_skel: one kernel per iteration does erode -> erode -> dilate+update
// through a 3-stage LDS pipeline (radius-3 halo). 37 launches total vs 102.
// CDNA5: async global->LDS halo loads (ASYNCcnt) + wave32 LDS stencil stages.
// ---------------------------------------------------------------------------

constexpr int D = 96, H = 128, W = 128;
constexpr size_t NVOX = (size_t)D * H * W;            // 1,572,864
constexpr int TD = 4, TH = 8, TW = 32;                // output tile per block

// LDS stage extents (halo radius 3 / 2 / 1 around the TDxTHxTW tile)
constexpr int IZ = TD + 6, IY = TH + 6, IX = TW + 6;  // 10x14x38 img (iter)
constexpr int AZ = TD + 4, AY = TH + 4, AX = TW + 4;  // 8x12x36  E1 (iter) / img (init)
constexpr int BZ = TD + 2, BY = TH + 2, BX = TW + 2;  // 6x10x34  E2 (iter) / E1 (init)
constexpr int NIMG = IZ * IY * IX;                    // 5320
constexpr int NA   = AZ * AY * AX;                    // 3456
constexpr int NB   = BZ * BY * BX;                    // 2040
constexpr int NRED_BLOCKS = 384;
constexpr int NSUMS = 10;

#if defined(__HIP_DEVICE_COMPILE__) && defined(__gfx1250__)
#if __has_builtin(__builtin_amdgcn_global_load_async_to_lds_b32)
#define USE_ASYNC_LDS 1
#endif
#endif

#ifdef USE_ASYNC_LDS
typedef __attribute__((address_space(1))) int g_i32;
typedef __attribute__((address_space(3))) int l_i32;
#endif

__device__ __forceinline__ int clampi(int v, int lo, int hi) {
  return min(max(v, lo), hi);
}

// Load a clamped (edge-replicated) halo tile of radius R into LDS via async loads.
// Clamp-to-edge == reference's -inf/+inf SAME padding for max/min pooling.
template <int R, int NZ, int NY, int NX>
__device__ __forceinline__ void load_halo(const float* __restrict__ src, float* tile,
                                          int z0, int y0, int x0, int tid) {
  constexpr int NCELL = NZ * NY * NX;
  for (int s = tid; s < NCELL; s += 256) {
    int lz = s / (NY * NX);
    int r  = s - lz * (NY * NX);
    int ly = r / NX;
    int lx = r - ly * NX;
    int gz = clampi(z0 + lz - R, 0, D - 1);
    int gy = clampi(y0 + ly - R, 0, H - 1);
    int gx = clampi(x0 + lx - R, 0, W - 1);
    size_t gi = ((size_t)gz * H + gy) * W + gx;
#ifdef USE_ASYNC_LDS
    __builtin_amdgcn_global_load_async_to_lds_b32(
        (g_i32*)(src + gi), (l_i32*)(&tile[s]), /*offset=*/0, /*cpol=*/0);
#else
    tile[s] = src[gi];
#endif
  }
#ifdef USE_ASYNC_LDS
#if __has_builtin(__builtin_amdgcn_s_wait_asynccnt)
  __builtin_amdgcn_s_wait_asynccnt(0);
#else
  asm volatile("s_wait_asynccnt 0" ::: "memory");
#endif
#endif
}

// One erosion stage in LDS. Output cells live at halo radius R; the input array
// has radius R+1. Halo cells are evaluated AT THEIR CLAMPED coordinate, which
// keeps clamp-to-edge semantics exact across fused stages. If gout!=null, the
// interior (radius-0) region is also streamed to global (the new img).
template <int R, int INY, int INX, int OZ, int OY, int OX>
__device__ __forceinline__ void erode_stage(const float* sin, float* sout,
                                            int z0, int y0, int x0, int tid,
                                            float* __restrict__ gout) {
  constexpr int NCELL = OZ * OY * OX;
  for (int s = tid; s < NCELL; s += 256) {
    int lz = s / (OY * OX);
    int r  = s - lz * (OY * OX);
    int ly = r / OX;
    int lx = r - ly * OX;
    int gz = z0 + lz - R, gy = y0 + ly - R, gx = x0 + lx - R;
    int iz = clampi(gz, 0, D - 1) - z0 + (R + 1);
    int iy = clampi(gy, 0, H - 1) - y0 + (R + 1);
    int ix = clampi(gx, 0, W - 1) - x0 + (R + 1);
    float m = 3.4e38f;
#pragma unroll
    for (int dz = -1; dz <= 1; ++dz)
#pragma unroll
      for (int dy = -1; dy <= 1; ++dy)
#pragma unroll
        for (int dx = -1; dx <= 1; ++dx)
          m = fminf(m, sin[((iz + dz) * INY + (iy + dy)) * INX + (ix + dx)]);
    sout[s] = m;
    if (gout && lz >= R && lz < R + TD && ly >= R && ly < R + TH &&
        lx >= R && lx < R + TW) {
      gout[((size_t)gz * H + gy) * W + gx] = m;
    }
  }
}

// Final stage: m = dilate3x3x3(sB) on the interior; d = relu(ref - m) with ref
// taken from the radius-2 LDS array; init or update skel in global memory.
// Each thread owns one (oy,ox) column: 6 plane-maxes feed all 4 z-outputs.
template <bool UPDATE>
__device__ __forceinline__ void dilate_out(const float* sB, const float* sRef,
                                           float* __restrict__ skel,
                                           int z0, int y0, int x0, int tid) {
  const int ox = tid & (TW - 1);
  const int oy = tid >> 5;
  float pm[BZ];
#pragma unroll
  for (int lz = 0; lz < BZ; ++lz) {
    const float* p = &sB[(lz * BY + oy) * BX + ox];
    float r0 = fmaxf(fmaxf(p[0], p[1]), p[2]);
    float r1 = fmaxf(fmaxf(p[BX], p[BX + 1]), p[BX + 2]);
    float r2 = fmaxf(fmaxf(p[2 * BX], p[2 * BX + 1]), p[2 * BX + 2]);
    pm[lz] = fmaxf(fmaxf(r0, r1), r2);
  }
#pragma unroll
  for (int oz = 0; oz < TD; ++oz) {
    float m = fmaxf(fmaxf(pm[oz], pm[oz + 1]), pm[oz + 2]);
    float refv = sRef[((oz + 2) * AY + (oy + 2)) * AX + (ox + 2)];
    float dd = fmaxf(refv - m, 0.0f);
    size_t gi = ((size_t)(z0 + oz) * H + (y0 + oy)) * W + (x0 + ox);
    if (UPDATE) {
      float s0 = skel[gi];
      skel[gi] = s0 + fmaxf(dd - s0 * dd, 0.0f);
    } else {
      skel[gi] = dd;
    }
  }
}

// One full soft_skel iteration:
//   img_out = erode(img_in)                                  (E1, written out)
//   d = relu(img_out - dilate(erode(img_out)))               (E2 via LDS)
//   skel += relu(d - skel*d)
__global__ __launch_bounds__(256) void skel_iter_kernel(const float* __restrict__ img_in,
                                                        float* __restrict__ img_out,
                                                        float* __restrict__ skel) {
  __shared__ float sImg[NIMG];
  __shared__ float sE1[NA];
  __shared__ float sE2[NB];
  const int tid = threadIdx.x;
  const int x0 = blockIdx.x * TW, y0 = blockIdx.y * TH, z0 = blockIdx.z * TD;

  load_halo<3, IZ, IY, IX>(img_in, sImg, z0, y0, x0, tid);
  __syncthreads();
  erode_stage<2, IY, IX, AZ, AY, AX>(sImg, sE1, z0, y0, x0, tid, img_out);
  __syncthreads();
  erode_stage<1, AY, AX, BZ, BY, BX>(sE1, sE2, z0, y0, x0, tid, nullptr);
  __syncthreads();
  dilate_out<true>(sE2, sE1, skel, z0, y0, x0, tid);
}

// Skeleton init: skel = relu(img - dilate(erode(img)))
__global__ __launch_bounds__(256) void skel_init_kernel(const float* __restrict__ img,
                                                        float* __restrict__ skel) {
  __shared__ float sImg[NA];
  __shared__ float sE1[NB];
  const int tid = threadIdx.x;
  const int x0 = blockIdx.x * TW, y0 = blockIdx.y * TH, z0 = blockIdx.z * TD;

  load_halo<2, AZ, AY, AX>(img, sImg, z0, y0, x0, tid);
  __syncthreads();
  erode_stage<1, AY, AX, BZ, BY, BX>(sImg, sE1, z0, y0, x0, tid, nullptr);
  __syncthreads();
  dilate_out<false>(sE1, sImg, skel, z0, y0, x0, tid);
}

// yp = sigmoid(pred); ypp = sigmoid(yp)   (the reference double-sigmoids)
__global__ __launch_bounds__(256) void sigmoid2_kernel(const float* __restrict__ pred,
                                                       float* __restrict__ yp,
                                                       float* __restrict__ ypp) {
  const float4* p4 = (const float4*)pred;
  float4* a4 = (float4*)yp;
  float4* b4 = (float4*)ypp;
  const int n4 = (int)(NVOX / 4);
  const int stride = gridDim.x * blockDim.x;
  for (int i = blockIdx.x * blockDim.x + threadIdx.x; i < n4; i += stride) {
    float4 v = p4[i];
    float4 a, b;
    a.x = 1.0f / (1.0f + expf(-v.x));
    a.y = 1.0f / (1.0f + expf(-v.y));
    a.z = 1.0f / (1.0f + expf(-v.z));
    a.w = 1.0f / (1.0f + expf(-v.w));
    b.x = 1.0f / (1.0f + expf(-a.x));
    b.y = 1.0f / (1.0f + expf(-a.y));
    b.z = 1.0f / (1.0f + expf(-a.z));
    b.w = 1.0f / (1.0f + expf(-a.w));
    a4[i] = a;
    b4[i] = b;
  }
}

// 10 partial sums per block, deterministic fixed-order LDS trees.
__global__ __launch_bounds__(256) void reduce_kernel(const float* __restrict__ yp,
                                                     const float* __restrict__ ypp,
                                                     const float* __restrict__ yt,
                                                     const float* __restrict__ sp,
                                                     const float* __restrict__ st,
                                                     float* __restrict__ partials) {
  float acc[NSUMS];
#pragma unroll
  for (int q = 0; q < NSUMS; ++q) acc[q] = 0.0f;
  const size_t stride = (size_t)gridDim.x * blockDim.x;
  for (size_t i = (size_t)blockIdx.x * blockDim.x + threadIdx.x; i < NVOX; i += stride) {
    float vp  = yp[i];
    float vpp = ypp[i];
    float vt  = yt[i];
    float vsp = sp[i];
    float vst = st[i];
    float w   = expf(-0.5f * vst * vst);   // SIGMA = 1
    acc[0] += vsp * vt;     // sum(skel_pred * y_true)
    acc[1] += vsp;          // sum(skel_pred)
    acc[2] += vst * vpp;    // sum(skel_true * ypp)
    acc[3] += vst;          // sum(skel_true)
    acc[4] += w * vp * vt;  // w_inter
    acc[5] += w * vp;       // w_pred
    acc[6] += w * vt;       // w_true
    acc[7] += vp * vt;      // inter
    acc[8] += vp;           // sum(yp)
    acc[9] += vt;           // sum(y_true)
  }
  __shared__ float smem[256];
  for (int q = 0; q < NSUMS; ++q) {
    smem[threadIdx.x] = acc[q];
    __syncthreads();
    for (int off = 128; off > 0; off >>= 1) {
      if (threadIdx.x < (unsigned)off) smem[threadIdx.x] += smem[threadIdx.x + off];
      __syncthreads();
    }
    if (threadIdx.x == 0) partials[q * gridDim.x + blockIdx.x] = smem[0];
    __syncthreads();
  }
}

__global__ __launch_bounds__(256) void finalize_kernel(const float* __restrict__ partials,
                                                       float* __restrict__ out) {
  __shared__ float smem[256];
  __shared__ float S[NSUMS];
  for (int q = 0; q < NSUMS; ++q) {
    float a = 0.0f;
    for (int i = threadIdx.x; i < NRED_BLOCKS; i += 256) a += partials[q * NRED_BLOCKS + i];
    smem[threadIdx.x] = a;
    __syncthreads();
    for (int off = 128; off > 0; off >>= 1) {
      if (threadIdx.x < (unsigned)off) smem[threadIdx.x] += smem[threadIdx.x + off];
      __syncthreads();
    }
    if (threadIdx.x == 0) S[q] = smem[0];
    __syncthreads();
  }
  if (threadIdx.x == 0) {
    const float SM = 1.0f;
    float tprec = (S[0] + SM) / (S[1] + SM);
    float tsens = (S[2] + SM) / (S[3] + SM);
    float cldice = 1.0f - 2.0f * tprec * tsens / (tprec + tsens + SM);
    float boundary = 1.0f - (2.0f * S[4] + SM) / (S[5] + S[6] + SM);
    float dice = 1.0f - (2.0f * S[7] + SM) / (S[8] + S[9] + SM);
    out[0] = 0.5f * cldice + 0.3f * boundary + 0.2f * dice;   // ALPHA=.5, BETA=.3
  }
}

// soft_skel: init + 16 fused iterations (img ping-pongs A/B; img0 stays intact).
static void soft_skel_enqueue(const float* img0, float* skel, float* A, float* B,
                              hipStream_t s) {
  dim3 g(W / TW, H / TH, D / TD), b(256);
  skel_init_kernel<<<g, b, 0, s>>>(img0, skel);
  const float* in = img0;
  for (int it = 0; it < 16; ++it) {
    float* out = (it & 1) ? B : A;
    skel_iter_kernel<<<g, b, 0, s>>>(in, out, skel);
    in = out;
  }
}

extern "C" void kernel_launch(void* const* d_in, const int* in_sizes, int n_in,
                              void* d_out, int out_size, void* d_ws, size_t ws_size,
                              hipStream_t stream) {
  const float* y_pred = (const float*)d_in[0];
  const float* y_true = (const float*)d_in[1];
  float* ws = (float*)d_ws;

  // Workspace layout: 6 volume buffers + partials (~38 MB total).
  float* yp       = ws + 0 * NVOX;
  float* ypp      = ws + 1 * NVOX;
  float* sp       = ws + 2 * NVOX;   // skel_pred
  float* st       = ws + 3 * NVOX;   // skel_true
  float* A        = ws + 4 * NVOX;
  float* B        = ws + 5 * NVOX;
  float* partials = ws + 6 * NVOX;   // NSUMS * NRED_BLOCKS floats

  sigmoid2_kernel<<<dim3(512), dim3(256), 0, stream>>>(y_pred, yp, ypp);
  soft_skel_enqueue(ypp, sp, A, B, stream);        // skel_pred = soft_skel(sigmoid(yp))
  soft_skel_enqueue(y_true, st, A, B, stream);     // skel_true = soft_skel(y_true)
  reduce_kernel<<<dim3(NRED_BLOCKS), dim3(256), 0, stream>>>(yp, ypp, y_true, sp, st, partials);
  finalize_kernel<<<dim3(1), dim3(256), 0, stream>>>(partials, (float*)d_out);
}